// GraphBackbone_13219909337662
// MI455X (gfx1250) — compile-verified
//
#include <hip/hip_runtime.h>
#include <stdint.h>

// Problem constants (match reference)
#define G_TOT   2048          // B*NW = 64*32 graphs
#define N_NODE  62
#define NTOT    (G_TOT * N_NODE)   // 126976 rows (multiple of 32)
#define CIN     128
#define HDIM    256
#define EDGES   496
#define LN_EPS  1e-5f

typedef __attribute__((ext_vector_type(2))) float v2f;
typedef __attribute__((ext_vector_type(8))) float v8f;

// ---------------------------------------------------------------------------
// Kernel 1: build per-graph CSR (by dst) + symmetric GCN normalization.
// Topology is identical for all 2048 graphs -> compute once (62 nodes / 496
// edges), single block.
// ---------------------------------------------------------------------------
__global__ void __launch_bounds__(64)
build_csr_kernel(const long long* __restrict__ ei,   // [2, E] int64 (src row 0, dst row 1)
                 int* __restrict__ ptr,              // [N+1]
                 int* __restrict__ csrc,             // [E]
                 float* __restrict__ cw,             // [E] edge norm, sorted by dst
                 float* __restrict__ selfw)          // [N] self-loop norm = 1/deg
{
    __shared__ int   cnt[N_NODE];
    __shared__ float dinv[N_NODE];
    __shared__ int   pfx[N_NODE + 1];
    const int t = threadIdx.x;

    if (t < N_NODE) cnt[t] = 1;            // self-loop contributes to degree
    __syncthreads();
    for (int e = t; e < EDGES; e += 64) {
        int d = (int)ei[EDGES + e];
        atomicAdd(&cnt[d], 1);
    }
    __syncthreads();
    if (t < N_NODE) dinv[t] = rsqrtf((float)cnt[t]);   // deg >= 1 always
    __syncthreads();
    if (t == 0) {                          // serial prefix over 62 entries
        int run = 0;
        for (int i = 0; i < N_NODE; ++i) { pfx[i] = run; run += cnt[i] - 1; }
        pfx[N_NODE] = run;                 // == EDGES
    }
    __syncthreads();
    if (t < N_NODE + 1) ptr[t] = pfx[t];
    if (t < N_NODE) { selfw[t] = dinv[t] * dinv[t]; cnt[t] = pfx[t]; }
    __syncthreads();
    for (int e = t; e < EDGES; e += 64) {
        int s = (int)ei[e];
        int d = (int)ei[EDGES + e];
        int pos = atomicAdd(&cnt[d], 1);
        csrc[pos] = s;
        cw[pos]   = dinv[s] * dinv[d];
    }
}

// ---------------------------------------------------------------------------
// Kernel 2: fp32 WMMA GEMM  out[NTOT,256] = A[NTOT,K] @ W[K,256]
// Block = 128 threads (4 waves). Block covers 32 rows x 256 cols; each wave
// owns a 32x64 slab = 8 accumulator tiles. Per 4-K step a wave loads two A
// fragments (b64 each) + four B fragments, and each B fragment feeds TWO
// WMMAs (row tiles share B) -> 8 independent v_wmma_f32_16x16x4_f32 per step,
// hiding load latency that previously stalled on s_wait_loadcnt.
// A-fragment layout per ISA 7.12.2: lanes 0-15 M=0..15, VGPR{0,1}=K{0,1};
// lanes 16-31 carry K{2,3}. B mirrored (lane half selects K pair, lane%16 = N).
// ---------------------------------------------------------------------------
__global__ void __launch_bounds__(128)
gemm_wmma_kernel(const float* __restrict__ A,
                 const float* __restrict__ W,     // [K, 256]
                 float* __restrict__ out,         // [NTOT, 256]
                 int K)
{
    const int row0 = blockIdx.x << 5;    // 32 rows per block
    const int lane = threadIdx.x & 31;
    const int wave = threadIdx.x >> 5;
    const int col0 = wave << 6;          // 4 waves * 64 cols = 256
    const int half = lane >> 4;          // 0: K+{0,1}  1: K+{2,3}
    const int l16  = lane & 15;

    const float* Arow0 = A + (size_t)(row0 + l16) * K;        // rows 0..15
    const float* Arow1 = Arow0 + (size_t)16 * K;              // rows 16..31

    v8f acc00 = {}, acc01 = {}, acc02 = {}, acc03 = {};
    v8f acc10 = {}, acc11 = {}, acc12 = {}, acc13 = {};

    for (int k = 0; k < K; k += 4) {
        if ((k & 63) == 0) {             // keep the A streams hot (global_prefetch)
            __builtin_prefetch(Arow0 + k + 64, 0, 1);
            __builtin_prefetch(Arow1 + k + 64, 0, 1);
        }

        v2f a0, a1;
        a0.x = Arow0[k + 2 * half + 0];
        a0.y = Arow0[k + 2 * half + 1];
        a1.x = Arow1[k + 2 * half + 0];
        a1.y = Arow1[k + 2 * half + 1];

        const float* Wk = W + (size_t)(k + 2 * half) * HDIM + col0 + l16;
        v2f b0, b1, b2, b3;
        b0.x = Wk[0];       b0.y = Wk[HDIM + 0];
        b1.x = Wk[16];      b1.y = Wk[HDIM + 16];
        b2.x = Wk[32];      b2.y = Wk[HDIM + 32];
        b3.x = Wk[48];      b3.y = Wk[HDIM + 48];

        acc00 = __builtin_amdgcn_wmma_f32_16x16x4_f32(false, a0, false, b0, (short)0, acc00, false, false);
        acc10 = __builtin_amdgcn_wmma_f32_16x16x4_f32(false, a1, false, b0, (short)0, acc10, false, false);
        acc01 = __builtin_amdgcn_wmma_f32_16x16x4_f32(false, a0, false, b1, (short)0, acc01, false, false);
        acc11 = __builtin_amdgcn_wmma_f32_16x16x4_f32(false, a1, false, b1, (short)0, acc11, false, false);
        acc02 = __builtin_amdgcn_wmma_f32_16x16x4_f32(false, a0, false, b2, (short)0, acc02, false, false);
        acc12 = __builtin_amdgcn_wmma_f32_16x16x4_f32(false, a1, false, b2, (short)0, acc12, false, false);
        acc03 = __builtin_amdgcn_wmma_f32_16x16x4_f32(false, a0, false, b3, (short)0, acc03, false, false);
        acc13 = __builtin_amdgcn_wmma_f32_16x16x4_f32(false, a1, false, b3, (short)0, acc13, false, false);
    }

    // C/D layout: VGPR r -> M = r + 8*half, N = lane%16 (+16*tile)
    float* Out0 = out + ((size_t)(row0 + 8 * half)) * HDIM + col0 + l16;
    float* Out1 = Out0 + (size_t)16 * HDIM;
#pragma unroll
    for (int r = 0; r < 8; ++r) {
        Out0[(size_t)r * HDIM +  0] = acc00[r];
        Out0[(size_t)r * HDIM + 16] = acc01[r];
        Out0[(size_t)r * HDIM + 32] = acc02[r];
        Out0[(size_t)r * HDIM + 48] = acc03[r];
        Out1[(size_t)r * HDIM +  0] = acc10[r];
        Out1[(size_t)r * HDIM + 16] = acc11[r];
        Out1[(size_t)r * HDIM + 32] = acc12[r];
        Out1[(size_t)r * HDIM + 48] = acc13[r];
    }
}

// ---------------------------------------------------------------------------
// Kernel 3: fused GCN aggregation + bias + LayerNorm + ReLU (+ residual).
// One block per node (126976 blocks), thread h = channel h.
// out[node,h] = relu(LN(selfw*xw[node] + sum_e cw[e]*xw[src_e] + b)) (+res)
// In-place safe when out == residual (each element read+written by its own
// thread only).
// ---------------------------------------------------------------------------
__global__ void __launch_bounds__(HDIM)
agg_ln_kernel(const float* __restrict__ XW,
              const float* __restrict__ bias,
              const float* __restrict__ gamma,
              const float* __restrict__ beta,
              const float* __restrict__ residual,   // nullptr for input layer
              float* __restrict__ out,
              const int* __restrict__ ptr,
              const int* __restrict__ csrc,
              const float* __restrict__ cw,
              const float* __restrict__ selfw)
{
    const int node = blockIdx.x;
    const int g    = node / N_NODE;
    const int n    = node - g * N_NODE;
    const int h    = threadIdx.x;
    const size_t gbase = (size_t)g * N_NODE;

    float acc = selfw[n] * XW[(size_t)node * HDIM + h];
    const int e0 = ptr[n], e1 = ptr[n + 1];
    for (int e = e0; e < e1; ++e) {
        int s = csrc[e];
        acc += cw[e] * XW[(gbase + s) * HDIM + h];
    }
    acc += bias[h];

    __shared__ float sm[HDIM];
    sm[h] = acc;
    __syncthreads();
#pragma unroll
    for (int s = HDIM / 2; s > 0; s >>= 1) {
        if (h < s) sm[h] += sm[h + s];
        __syncthreads();
    }
    const float mean = sm[0] * (1.0f / HDIM);
    __syncthreads();

    const float d = acc - mean;
    sm[h] = d * d;
    __syncthreads();
#pragma unroll
    for (int s = HDIM / 2; s > 0; s >>= 1) {
        if (h < s) sm[h] += sm[h + s];
        __syncthreads();
    }
    const float var = sm[0] * (1.0f / HDIM);

    float y = d * rsqrtf(var + LN_EPS) * gamma[h] + beta[h];
    y = fmaxf(y, 0.0f);
    if (residual) y += residual[(size_t)node * HDIM + h];
    out[(size_t)node * HDIM + h] = y;
}

// ---------------------------------------------------------------------------
// Launcher
// ---------------------------------------------------------------------------
extern "C" void kernel_launch(void* const* d_in, const int* in_sizes, int n_in,
                              void* d_out, int out_size, void* d_ws, size_t ws_size,
                              hipStream_t stream)
{
    (void)in_sizes; (void)n_in; (void)out_size; (void)ws_size;

    const float*     x        = (const float*)d_in[0];
    /* d_in[1] base_adj unused (conv_type='gcn') */
    const long long* ei       = (const long long*)d_in[2];   // int64 [2,E]
    const float*     W_in     = (const float*)d_in[3];
    const float*     b_in     = (const float*)d_in[4];
    const float*     W_hid    = (const float*)d_in[5];       // [3,256,256]
    const float*     b_hid    = (const float*)d_in[6];       // [3,256]
    const float*     g_in     = (const float*)d_in[7];
    const float*     beta_in  = (const float*)d_in[8];
    const float*     g_hid    = (const float*)d_in[9];       // [3,256]
    const float*     beta_hid = (const float*)d_in[10];      // [3,256]
    float*           out      = (float*)d_out;

    // Workspace layout
    char*  ws    = (char*)d_ws;
    int*   ptr   = (int*)(ws);                    //  (N+1) ints
    int*   csrc  = (int*)(ws + 256);              //  E ints
    float* cw    = (float*)(ws + 256 + 2048);     //  E floats
    float* selfw = (float*)(ws + 256 + 4096);     //  N floats
    float* bufG  = (float*)(ws + 8192);                       // 130 MB GEMM out
    float* act   = bufG + (size_t)NTOT * HDIM;                // 130 MB activations

    build_csr_kernel<<<1, 64, 0, stream>>>(ei, ptr, csrc, cw, selfw);

    // Input layer: 128 -> 256
    gemm_wmma_kernel<<<NTOT / 32, 128, 0, stream>>>(x, W_in, bufG, CIN);
    agg_ln_kernel<<<NTOT, HDIM, 0, stream>>>(bufG, b_in, g_in, beta_in,
                                             /*residual=*/nullptr, act,
                                             ptr, csrc, cw, selfw);

    // Hidden layers: 256 -> 256, residual; layer 2 writes d_out
    for (int l = 0; l < 3; ++l) {
        gemm_wmma_kernel<<<NTOT / 32, 128, 0, stream>>>(
            act, W_hid + (size_t)l * HDIM * HDIM, bufG, HDIM);
        float* dst = (l == 2) ? out : act;
        agg_ln_kernel<<<NTOT, HDIM, 0, stream>>>(
            bufG, b_hid + (size_t)l * HDIM,
            g_hid + (size_t)l * HDIM, beta_hid + (size_t)l * HDIM,
            /*residual=*/act, dst, ptr, csrc, cw, selfw);
    }
}